// BrushStroke_49074296324493
// MI455X (gfx1250) — compile-verified
//
#include <hip/hip_runtime.h>
#include <hip/hip_bf16.h>

typedef float v2f __attribute__((ext_vector_type(2)));
typedef float v8f __attribute__((ext_vector_type(8)));

#define IMAGE_W   256
#define IMAGE_H   256
#define PATCH     16
#define PAD       8
#define GRID_PAD  (IMAGE_W + 2 * PAD)      // 272 coords in the normalization sum
#define INV2SIG2  12.5f                    // 1/(2*0.2^2)
#define EPSF      1e-7f
#define NBRUSH    64
#define HSLAB     64                       // rows of output per workgroup

// One workgroup = (b, c, h-slab). 8 waves; each wave accumulates 8 fp32 16x16
// output tiles across the n-loop with v_wmma_f32_16x16x4_f32.
__global__ __launch_bounds__(256)
void BrushStroke_wmma_kernel(const float* __restrict__ brushes,
                             const float* __restrict__ patches,
                             float* __restrict__ out) {
    const int blk   = blockIdx.x;
    const int hs    = blk & 3;             // which 64-row slab
    const int c     = (blk >> 2) % 3;
    const int b     = blk / 12;
    const int hBase = hs * HSLAB;

    __shared__ float sFx[IMAGE_W * PATCH]; // [w][q]  16 KB
    __shared__ float sT [PATCH * IMAGE_W]; // [p][w]  16 KB  (t transposed: B operand)
    __shared__ float sFy[HSLAB * PATCH];   // [h_local][p] 4 KB (A operand)
    __shared__ float sP [PATCH * PATCH];   // patch [p][q]
    __shared__ float sDen[2 * PATCH * 16]; // partial denom sums [filter][q][chunk]
    __shared__ float sInvDenX[PATCH];
    __shared__ float sInvDenY[PATCH];

    const int tid  = threadIdx.x;
    const int lane = tid & 31;
    const int wave = tid >> 5;

    // WMMA fp32 lane mapping (wave32):
    //   A 16x4 : lane L slot s -> m = L&15, k = 2*(L>>4) + s
    //   B 4x16 : lane L slot s -> n = L&15, k = 2*(L>>4) + s
    //   D 16x16: vgpr v, lane L -> m = v + 8*(L>>4), n = L&15
    const int mIdx     = lane & 15;
    const int kBaseSel = (lane >> 4) << 1; // 0 or 2

    v8f acc[8];
#pragma unroll
    for (int i = 0; i < 8; ++i) acc[i] = (v8f)0.0f;

    for (int n = 0; n < NBRUSH; ++n) {
        const float* Pg = patches + ((((size_t)b * NBRUSH + n) * 3 + c) * (PATCH * PATCH));
        // Stage patch tile for this (b,n,c); prefetch the next one (global_prefetch_b8).
        sP[tid] = Pg[tid];
        if (n + 1 < NBRUSH) {
            __builtin_prefetch(Pg + 3 * PATCH * PATCH, 0, 0);
        }

        const float gx = brushes[((size_t)b * NBRUSH + n) * 2 + 0] * (float)IMAGE_W;
        const float gy = brushes[((size_t)b * NBRUSH + n) * 2 + 1] * (float)IMAGE_H;

        // ---- filter denominators: sum exp over all 272 padded coords, per q ----
        {
            const int q     = tid & 15;
            const int chunk = tid >> 4;          // 16 chunks x 17 coords = 272
            const float muX = gx + (float)q - 7.5f;
            const float muY = gy + (float)q - 7.5f;
            float sx = 0.0f, sy = 0.0f;
            const int i0 = chunk * 17;
#pragma unroll
            for (int i = 0; i < 17; ++i) {
                const float coord = (float)(i0 + i - PAD);
                const float dx = coord - muX;
                const float dy = coord - muY;
                sx += __expf(-dx * dx * INV2SIG2);
                sy += __expf(-dy * dy * INV2SIG2);
            }
            sDen[q * 16 + chunk]       = sx;
            sDen[256 + q * 16 + chunk] = sy;
        }
        __syncthreads();
        if (tid < 32) {
            const int f = tid >> 4;
            const int q = tid & 15;
            float s = 0.0f;
#pragma unroll
            for (int k = 0; k < 16; ++k) s += sDen[f * 256 + q * 16 + k];
            const float inv = 1.0f / (s + EPSF);
            if (f == 0) sInvDenX[q] = inv; else sInvDenY[q] = inv;
        }
        __syncthreads();

        // ---- Fx[w][q] (one w per thread) and Fy[h][p] slab ----
        {
            const int w = tid;
            const float mu0 = gx - 7.5f;
#pragma unroll
            for (int q = 0; q < PATCH; ++q) {
                const float d = (float)w - (mu0 + (float)q);
                sFx[w * PATCH + q] = __expf(-d * d * INV2SIG2) * sInvDenX[q];
            }
        }
#pragma unroll
        for (int idx = tid; idx < HSLAB * PATCH; idx += 256) {
            const int hl = idx >> 4;
            const int p  = idx & 15;
            const float d = (float)(hBase + hl) - (gy + (float)p - 7.5f);
            sFy[idx] = __expf(-d * d * INV2SIG2) * sInvDenY[p];
        }
        __syncthreads();

        // ---- tCol[p][w] = sum_q P[p][q] * Fx[w][q]  (t transposed, B operand) ----
        {
            const int w = tid;
            float fx[PATCH];
#pragma unroll
            for (int q = 0; q < PATCH; ++q) fx[q] = sFx[w * PATCH + q];
#pragma unroll
            for (int p = 0; p < PATCH; ++p) {
                float s = 0.0f;
#pragma unroll
                for (int q = 0; q < PATCH; ++q) s += sP[p * PATCH + q] * fx[q];
                sT[p * IMAGE_W + w] = s;
            }
        }
        __syncthreads();

        // ---- WMMA accumulate: out_tile(h,w) += Fy_tile[16x16p] x tCol_tile[16p x 16w]
        {
            const int hTile = wave >> 1;          // 0..3
            const int wT0   = (wave & 1) * 8;     // 8 w-tiles per wave
#pragma unroll
            for (int t8 = 0; t8 < 8; ++t8) {
                const int wTile = wT0 + t8;
                const int colB  = wTile * 16 + mIdx;
                const int rowA  = (hTile * 16 + mIdx) * PATCH;
#pragma unroll
                for (int p0 = 0; p0 < PATCH; p0 += 4) {
                    v2f A, Bv;
                    A.x  = sFy[rowA + p0 + kBaseSel];
                    A.y  = sFy[rowA + p0 + kBaseSel + 1];
                    Bv.x = sT[(p0 + kBaseSel) * IMAGE_W + colB];
                    Bv.y = sT[(p0 + kBaseSel + 1) * IMAGE_W + colB];
                    acc[t8] = __builtin_amdgcn_wmma_f32_16x16x4_f32(
                        false, A, false, Bv, (short)0, acc[t8], false, false);
                }
            }
        }
        __syncthreads();
    }

    // ---- store: out[b][c][h][w] = acc * (1/N) ----
    {
        const int hTile = wave >> 1;
        const int wT0   = (wave & 1) * 8;
        const int mHi   = (lane >> 4) * 8;
        const float scale = 1.0f / (float)NBRUSH;
        float* outBase = out + (((size_t)b * 3 + c) * IMAGE_H) * IMAGE_W;
#pragma unroll
        for (int t8 = 0; t8 < 8; ++t8) {
            const int w = (wT0 + t8) * 16 + mIdx;
#pragma unroll
            for (int v = 0; v < 8; ++v) {
                const int h = hBase + hTile * 16 + mHi + v;
                outBase[(size_t)h * IMAGE_W + w] = acc[t8][v] * scale;
            }
        }
    }
}

extern "C" void kernel_launch(void* const* d_in, const int* in_sizes, int n_in,
                              void* d_out, int out_size, void* d_ws, size_t ws_size,
                              hipStream_t stream) {
    const float* brushes = (const float*)d_in[0]; // [32,64,2]
    const float* patches = (const float*)d_in[1]; // [32,64,3,16,16]
    float* out = (float*)d_out;                   // [32,3,256,256]
    (void)in_sizes; (void)n_in; (void)out_size; (void)d_ws; (void)ws_size;

    dim3 grid(32 * 3 * 4); // (b, c, h-slab)
    dim3 block(256);       // 8 waves of 32
    BrushStroke_wmma_kernel<<<grid, block, 0, stream>>>(brushes, patches, out);
}